// st_conv_block_50259707298087
// MI455X (gfx1250) — compile-verified
//
#include <hip/hip_runtime.h>
#include <math.h>

// Problem constants (from reference)
#define B_   8
#define C0_  16
#define C1_  64
#define C2_  64
#define T0_  12
#define T1_  10   // after conv1 (VALID, KT=3)
#define T2_  8    // after conv2
#define NN_  256
#define KT_  3
#define HH_  8
#define ALPHA_ 0.2f

typedef float v2f __attribute__((ext_vector_type(2)));
typedef float v4f __attribute__((ext_vector_type(4)));
typedef float v8f __attribute__((ext_vector_type(8)));

// Branch-free tanh: exact at asymptotes, ~1e-7 abs error near 0 (no EXEC dance)
__device__ __forceinline__ float tanh_fast(float x) {
  float e = __expf(fminf(2.0f * x, 80.0f));
  return (e - 1.0f) / (e + 1.0f);
}

// ---------------------------------------------------------------------------
// K0: u1[h][c] = sum_d W[h][c][d]*a1[h][d] ; u2[h][c] = sum_d W[h][c][d]*a2[h][d]
// ---------------------------------------------------------------------------
__global__ __launch_bounds__(256) void prep_u_kernel(
    const float* __restrict__ W,   // (H, C1, C1)
    const float* __restrict__ a,   // (H, 2*C1)
    float* __restrict__ u1, float* __restrict__ u2)
{
  int tid = blockIdx.x * 256 + threadIdx.x;
  if (tid >= HH_ * C1_) return;
  int h = tid >> 6, c = tid & 63;
  const float* Wh = W + ((size_t)h * C1_ + c) * C1_;
  const float* ah = a + (size_t)h * 2 * C1_;
  float s1 = 0.f, s2 = 0.f;
  for (int d = 0; d < C1_; ++d) {
    float w = Wh[d];
    s1 += w * ah[d];
    s2 += w * ah[C1_ + d];
  }
  u1[tid] = s1;
  u2[tid] = s2;
}

// ---------------------------------------------------------------------------
// K1: conv1 + GLU + padded residual -> xg in (b, t, n, c) layout
// ---------------------------------------------------------------------------
__global__ __launch_bounds__(256) void conv1_glu_kernel(
    const float* __restrict__ x,   // (B, C0, T0, N)
    const float* __restrict__ w,   // (2*C1, C0, KT, 1)
    const float* __restrict__ b,   // (2*C1)
    float* __restrict__ xg)        // (B, T1, N, C1)
{
  int idx = blockIdx.x * 256 + threadIdx.x;        // B*T1*N*C1 total
  int c  = idx & 63;
  int n  = (idx >> 6) & 255;
  int bt = idx >> 14;                              // 0..79
  int bb = bt / T1_, t = bt - bb * T1_;

  float accp = b[c];
  float accq = b[c + C1_];
  #pragma unroll
  for (int kt = 0; kt < KT_; ++kt) {
    for (int ci = 0; ci < C0_; ++ci) {
      float xv = x[(((size_t)bb * C0_ + ci) * T0_ + (t + kt)) * NN_ + n];
      accp = fmaf(w[((size_t)c         * C0_ + ci) * KT_ + kt], xv, accp);
      accq = fmaf(w[((size_t)(c + C1_) * C0_ + ci) * KT_ + kt], xv, accq);
    }
  }
  float xin = (c < C0_) ? x[(((size_t)bb * C0_ + c) * T0_ + (t + KT_ - 1)) * NN_ + n] : 0.f;
  float sg  = 1.f / (1.f + __expf(-accq));
  xg[idx]   = (accp + xin) * sg;
}

// ---------------------------------------------------------------------------
// K2: fused GAT attention per (b,t): softmax(masked leaky(e1_i+e2_j)) @ xg
//     via V_WMMA_F32_16X16X4_F32, tanh, mean over heads, residual -> x_s
//
// 256 threads = 8 waves; wave w owns row blocks {w, w+8} (16 rows each).
// - Softmax pass 2 stores UNNORMALIZED probs to a wave-private LDS tile;
//   WMMA K-loop reads A with one ds_load_b64 (PSTR=260 -> all-64-bank).
// - xg is staged twice: row-major (e-vectors/residual, XSTR=72) and
//   K-pair-interleaved (XPSTR=160, ==32 mod 64) so each B fragment is ONE
//   naturally packed, conflict-free ds_load_b64 (no repacking movs).
// - 1/rowsum applied once to the final accumulator (softmax is linear in it).
// ---------------------------------------------------------------------------
#define XSTR  72
#define XPSTR 160
#define PSTR  260

__global__ __launch_bounds__(256) void attn_kernel(
    const float* __restrict__ xg,   // (B, T1, N, C1)
    const float* __restrict__ adj,  // (N, N)
    const float* __restrict__ u1,   // (H, C1)
    const float* __restrict__ u2,   // (H, C1)
    float* __restrict__ xs)         // (B, T1, N, C1)
{
  extern __shared__ float smem[];
  float*    s_xg   = smem;                         // 256*72   = 18432
  float*    s_xgp  = s_xg + NN_ * XSTR;            // 128*160  = 20480
  float*    s_e1   = s_xgp + (NN_ / 2) * XPSTR;    // 256
  float*    s_e2   = s_e1 + NN_;                   // 256
  float*    s_rinv = s_e2 + NN_;                   // 256 (1/rowsum per row)
  float*    s_p    = s_rinv + NN_;                 // 8 waves * 16 * PSTR
  unsigned* s_adj  = (unsigned*)(s_p + 8 * 16 * PSTR); // 256*8 words

  const int bt   = blockIdx.x;
  const int tid  = threadIdx.x;
  const int lane = tid & 31;
  const int wave = tid >> 5;
  const int mloc = lane & 15;
  const int half = lane >> 4;

  const float* xg_bt = xg + (size_t)bt * NN_ * C1_;

  // Stage xg: row-major + K-pair-interleaved copies
  for (int i = tid; i < NN_ * C1_; i += 256) {
    int n = i >> 6, c = i & 63;
    float v = xg_bt[i];
    s_xg[n * XSTR + c] = v;
    s_xgp[(n >> 1) * XPSTR + 2 * c + (n & 1)] = v;
  }
  // Adjacency bitmask: row n = tid
  {
    int n = tid;
    for (int w8 = 0; w8 < 8; ++w8) {
      unsigned bits = 0u;
      for (int j = 0; j < 32; ++j)
        bits |= (adj[(size_t)n * NN_ + w8 * 32 + j] > 0.f) ? (1u << j) : 0u;
      s_adj[n * 8 + w8] = bits;
    }
  }
  __syncthreads();

  const float inv_h = 1.0f / (float)HH_;
  float* s_pw = s_p + wave * 16 * PSTR;           // wave-private prob tile

  // running mean-over-heads accumulators (registers, fixed ownership)
  float racc[2][4][8];
  #pragma unroll
  for (int i = 0; i < 2; ++i)
    #pragma unroll
    for (int j = 0; j < 4; ++j)
      #pragma unroll
      for (int r = 0; r < 8; ++r) racc[i][j][r] = 0.f;

  for (int h = 0; h < HH_; ++h) {
    // e1[n], e2[n] = xg[n,:] . u{1,2}[h,:]
    {
      int n = tid;
      const float* row = &s_xg[n * XSTR];
      const float* U1 = u1 + h * C1_;
      const float* U2 = u2 + h * C1_;
      float a1 = 0.f, a2 = 0.f;
      for (int c = 0; c < C1_; ++c) {
        float v = row[c];
        a1 = fmaf(v, U1[c], a1);
        a2 = fmaf(v, U2[c], a2);
      }
      s_e1[n] = a1;
      s_e2[n] = a2;
    }
    __syncthreads();

    #pragma unroll
    for (int rbi = 0; rbi < 2; ++rbi) {
      const int rb = wave + rbi * 8;       // row block 0..15
      const int m  = rb * 16 + mloc;       // my attention row
      const float e1m = s_e1[m];
      const int jbase = half * 128;        // each lane-half scans 128 cols

      // pass 1: row max (vectorized e2, adjacency word hoisted per 32 cols)
      float mx = -3.0e38f;
      for (int w8 = 0; w8 < 4; ++w8) {
        unsigned adjw = s_adj[m * 8 + half * 4 + w8];
        const float* eb = &s_e2[jbase + w8 * 32];
        for (int j4 = 0; j4 < 32; j4 += 4) {
          v4f e = *(const v4f*)(eb + j4);
          #pragma unroll
          for (int u = 0; u < 4; ++u) {
            float sc = e1m + e[u];
            sc = sc > 0.f ? sc : ALPHA_ * sc;
            int on = (adjw >> (j4 + u)) & 1;
            mx = fmaxf(mx, on ? sc : -3.0e38f);
          }
        }
      }
      mx = fmaxf(mx, __shfl_xor(mx, 16));

      // pass 2: sum of exp + store unnormalized probs (mask folded into exp arg)
      float sm = 0.f;
      for (int w8 = 0; w8 < 4; ++w8) {
        unsigned adjw = s_adj[m * 8 + half * 4 + w8];
        const float* eb = &s_e2[jbase + w8 * 32];
        float* pb = &s_pw[mloc * PSTR + jbase + w8 * 32];
        for (int j4 = 0; j4 < 32; j4 += 4) {
          v4f e = *(const v4f*)(eb + j4);
          v4f p;
          #pragma unroll
          for (int u = 0; u < 4; ++u) {
            float sc = e1m + e[u];
            sc = sc > 0.f ? sc : ALPHA_ * sc;
            sc = ((adjw >> (j4 + u)) & 1) ? sc : -3.0e38f;  // exp -> exact 0
            float pe = __expf(sc - mx);
            p[u] = pe;
            sm += pe;
          }
          *(v4f*)(pb + j4) = p;
        }
      }
      sm += __shfl_xor(sm, 16);
      s_rinv[rb * 16 + mloc] = 1.0f / sm;  // both halves write same value

      // 16x64 tile: 4 f32 WMMA accumulators, K over 256 in steps of 4
      v8f acc0 = {}, acc1 = {}, acc2 = {}, acc3 = {};
      const float* pa   = &s_pw[mloc * PSTR + 2 * half];
      const float* xpb  = &s_xgp[half * XPSTR + 2 * mloc];
      for (int k0 = 0; k0 < NN_; k0 += 4) {
        v2f a = *(const v2f*)(pa + k0);                  // one ds_load_b64
        const float* bp = xpb + (k0 >> 1) * XPSTR;       // rows {ka, ka+1} packed
        v2f b0 = *(const v2f*)(bp + 0);
        v2f b1 = *(const v2f*)(bp + 32);
        v2f b2 = *(const v2f*)(bp + 64);
        v2f b3 = *(const v2f*)(bp + 96);
        acc0 = __builtin_amdgcn_wmma_f32_16x16x4_f32(false, a, false, b0, (short)0, acc0, false, false);
        acc1 = __builtin_amdgcn_wmma_f32_16x16x4_f32(false, a, false, b1, (short)0, acc1, false, false);
        acc2 = __builtin_amdgcn_wmma_f32_16x16x4_f32(false, a, false, b2, (short)0, acc2, false, false);
        acc3 = __builtin_amdgcn_wmma_f32_16x16x4_f32(false, a, false, b3, (short)0, acc3, false, false);
      }

      // D layout: VGPR r -> M = rb*16 + r + 8*half, N = ntile*16 + mloc.
      // Apply softmax normalization here (linear), then tanh, then mean.
      #pragma unroll
      for (int r = 0; r < 8; ++r) {
        float inv = s_rinv[rb * 16 + r + 8 * half];
        racc[rbi][0][r] += tanh_fast(acc0[r] * inv) * inv_h;
        racc[rbi][1][r] += tanh_fast(acc1[r] * inv) * inv_h;
        racc[rbi][2][r] += tanh_fast(acc2[r] * inv) * inv_h;
        racc[rbi][3][r] += tanh_fast(acc3[r] * inv) * inv_h;
      }
    }
    __syncthreads();   // protect s_e1/s_e2 reuse next head
  }

  // x_s = x_t1 + mean_h tanh(head_out)
  float* xs_bt = xs + (size_t)bt * NN_ * C1_;
  #pragma unroll
  for (int rbi = 0; rbi < 2; ++rbi) {
    const int rb = wave + rbi * 8;
    #pragma unroll
    for (int r = 0; r < 8; ++r) {
      int mm = rb * 16 + r + 8 * half;
      const float* xrow = &s_xg[mm * XSTR + mloc];
      float* orow = xs_bt + (size_t)mm * C1_ + mloc;
      orow[0]  = xrow[0]  + racc[rbi][0][r];
      orow[16] = xrow[16] + racc[rbi][1][r];
      orow[32] = xrow[32] + racc[rbi][2][r];
      orow[48] = xrow[48] + racc[rbi][3][r];
    }
  }
}

// ---------------------------------------------------------------------------
// K3: conv2 + ReLU residual + LayerNorm over (N,C2) + transposed store
// one workgroup per (b, t2): 64 blocks x 256 threads; thread owns node n=tid
// ---------------------------------------------------------------------------
__global__ __launch_bounds__(256) void conv2_ln_kernel(
    const float* __restrict__ xs,   // (B, T1, N, C1)
    const float* __restrict__ w2,   // (C2, C1, KT, 1)
    const float* __restrict__ b2,   // (C2)
    const float* __restrict__ lnw,  // (N, C2)
    const float* __restrict__ lnb,  // (N, C2)
    float* __restrict__ out)        // (B, C2, T2, N)
{
  extern __shared__ float smem[];
  float* s_w   = smem;                       // 64*64*3 = 12288
  float* s_y   = s_w + C2_ * C1_ * KT_;      // 256*64  = 16384
  float* s_red = s_y + NN_ * C2_;            // 16

  const int bt = blockIdx.x;
  const int bb = bt >> 3, t = bt & 7;
  const int tid = threadIdx.x;

  for (int i = tid; i < C2_ * C1_ * KT_; i += 256) s_w[i] = w2[i];
  __syncthreads();

  const int n = tid;
  const float* xsb = xs + (((size_t)bb * T1_ + t) * NN_) * C1_;

  float lsum = 0.f, lsq = 0.f;
  for (int cb = 0; cb < 4; ++cb) {
    const int c0 = cb * 16;
    float acc[16];
    #pragma unroll
    for (int u = 0; u < 16; ++u) acc[u] = b2[c0 + u];
    for (int kt = 0; kt < KT_; ++kt) {
      const float* xrow = xsb + (size_t)kt * NN_ * C1_ + (size_t)n * C1_;
      for (int ci = 0; ci < C1_; ++ci) {
        float xv = xrow[ci];
        #pragma unroll
        for (int u = 0; u < 16; ++u)
          acc[u] = fmaf(s_w[((c0 + u) * C1_ + ci) * KT_ + kt], xv, acc[u]);
      }
    }
    const float* xres = xsb + (size_t)(KT_ - 1) * NN_ * C1_ + (size_t)n * C1_;
    #pragma unroll
    for (int u = 0; u < 16; ++u) {
      float v = acc[u] + xres[c0 + u];
      v = v > 0.f ? v : 0.f;
      s_y[n * C2_ + c0 + u] = v;
      lsum += v;
      lsq  += v * v;
    }
  }

  // block reduction of sum / sumsq over 256 threads (wave32 -> 8 partials)
  for (int off = 16; off > 0; off >>= 1) {
    lsum += __shfl_xor(lsum, off);
    lsq  += __shfl_xor(lsq,  off);
  }
  if ((tid & 31) == 0) { s_red[tid >> 5] = lsum; s_red[8 + (tid >> 5)] = lsq; }
  __syncthreads();
  float tsum = 0.f, tsq = 0.f;
  for (int i = 0; i < 8; ++i) { tsum += s_red[i]; tsq += s_red[8 + i]; }
  const float cnt  = (float)(NN_ * C2_);
  const float mu   = tsum / cnt;
  const float var  = tsq / cnt - mu * mu;
  const float rstd = rsqrtf(var + 1e-5f);

  for (int c = 0; c < C2_; ++c) {
    float v = (s_y[n * C2_ + c] - mu) * rstd * lnw[(size_t)n * C2_ + c] + lnb[(size_t)n * C2_ + c];
    out[(((size_t)bb * C2_ + c) * T2_ + t) * NN_ + n] = v;
  }
}

// ---------------------------------------------------------------------------
extern "C" void kernel_launch(void* const* d_in, const int* in_sizes, int n_in,
                              void* d_out, int out_size, void* d_ws, size_t ws_size,
                              hipStream_t stream) {
  const float* x   = (const float*)d_in[0];
  const float* adj = (const float*)d_in[1];
  const float* w1  = (const float*)d_in[2];
  const float* b1  = (const float*)d_in[3];
  const float* w2  = (const float*)d_in[4];
  const float* b2  = (const float*)d_in[5];
  const float* Wh  = (const float*)d_in[6];
  const float* ah  = (const float*)d_in[7];
  const float* lnw = (const float*)d_in[8];
  const float* lnb = (const float*)d_in[9];
  float* out = (float*)d_out;

  // workspace layout (floats)
  float* ws = (float*)d_ws;
  const size_t xg_elems = (size_t)B_ * T1_ * NN_ * C1_;   // 1,310,720
  float* xg = ws;
  float* xs = xg + xg_elems;
  float* u1 = xs + xg_elems;
  float* u2 = u1 + HH_ * C1_;

  prep_u_kernel<<<2, 256, 0, stream>>>(Wh, ah, u1, u2);

  const int n_c1 = (int)(xg_elems / 256);                 // 5120 blocks
  conv1_glu_kernel<<<n_c1, 256, 0, stream>>>(x, w1, b1, xg);

  const size_t attn_lds =
      (size_t)(NN_ * XSTR + (NN_ / 2) * XPSTR + 3 * NN_ + 8 * 16 * PSTR) * sizeof(float)
      + (size_t)(NN_ * 8) * sizeof(unsigned);             // ~293 KB
  attn_kernel<<<B_ * T1_, 256, attn_lds, stream>>>(xg, adj, u1, u2, xs);

  const size_t c2_lds = (size_t)(C2_ * C1_ * KT_ + NN_ * C2_ + 16) * sizeof(float); // ~112 KB
  conv2_ln_kernel<<<B_ * T2_, 256, c2_lds, stream>>>(xs, w2, b2, lnw, lnb, out);
}